// CSSA_47364899340391
// MI455X (gfx1250) — compile-verified
//
#include <hip/hip_runtime.h>
#include <stdint.h>

typedef float v2f __attribute__((ext_vector_type(2)));
typedef float v8f __attribute__((ext_vector_type(8)));

#define B_ 32
#define C_ 512
#define D_ 4096
#define H_ 128

// ---------------- Kernel 1: mean-pool over d ----------------
// One block (256 threads) per (b,c) row of 4096 floats; float4 coalesced loads.
__global__ void CSSA_pool_kernel(const float* __restrict__ x, float* __restrict__ pooled) {
    __shared__ float red[256];
    const int row = blockIdx.x;            // 0 .. B_*C_-1
    const int tid = threadIdx.x;
    const float4* xr = (const float4*)(x + (size_t)row * D_);
    float s = 0.f;
#pragma unroll
    for (int i = 0; i < 4; ++i) {
        float4 v = xr[tid + 256 * i];
        s += v.x + v.y + v.z + v.w;
    }
    red[tid] = s;
    __syncthreads();
    for (int off = 128; off > 0; off >>= 1) {
        if (tid < off) red[tid] += red[tid + off];
        __syncthreads();
    }
    if (tid == 0) pooled[row] = red[0] * (1.0f / (float)D_);
}

// ---------------- Kernel 2: tiny MLP via f32 WMMA ----------------
// scores[32,512] = LeakyReLU(pooled[32,512] @ W1^T + b1) @ W2^T + b2
// (sigmoid omitted: strictly monotonic, argsort ordering unchanged)
// One workgroup, 512 threads = 16 waves. Each wave owns 16x16 output tiles.
__global__ void CSSA_mlp_kernel(const float* __restrict__ pooled,
                                const float* __restrict__ W1, const float* __restrict__ b1,
                                const float* __restrict__ W2, const float* __restrict__ b2,
                                float* __restrict__ scores) {
    __shared__ float hbuf[B_ * H_];        // 32*128 floats = 16 KB
    const int tid  = threadIdx.x;
    const int wave = tid >> 5;
    const int lane = tid & 31;
    const int l15  = lane & 15;            // M index (A) / N index (B,C,D) within tile
    const int hi   = lane >> 4;            // 0 or 1
    const int kb   = hi << 1;              // K sub-offset: 0 or 2

    // ---- GEMM1: h = pooled @ W1^T ; 2x8 tiles of 16x16, one per wave; K=512
    {
        const int mbase = (wave >> 3) * 16;     // 0 or 16
        const int nbase = (wave & 7) * 16;      // 0..112
        v8f acc = {};
        const float* arow = pooled + (size_t)(mbase + l15) * C_;  // A lane row (M)
        const float* brow = W1     + (size_t)(nbase + l15) * C_;  // B^T lane row (N)
        for (int k0 = 0; k0 < C_; k0 += 4) {
            v2f a = *(const v2f*)(arow + k0 + kb);
            v2f b = *(const v2f*)(brow + k0 + kb);
            acc = __builtin_amdgcn_wmma_f32_16x16x4_f32(
                false, a, false, b, (short)0, acc, false, false);
        }
        const float bias = b1[nbase + l15];
#pragma unroll
        for (int r = 0; r < 8; ++r) {           // elem r -> M = r + 8*hi, N = l15
            float v = acc[r] + bias;
            v = (v >= 0.f) ? v : 0.01f * v;     // LeakyReLU(0.01)
            hbuf[(mbase + r + 8 * hi) * H_ + (nbase + l15)] = v;
        }
    }
    __syncthreads();

    // ---- GEMM2: scores = h @ W2^T ; 2x32 tiles, 4 per wave; K=128
    for (int i = 0; i < 4; ++i) {
        const int t = wave * 4 + i;             // 0..63
        const int mbase = (t >> 5) * 16;
        const int nbase = (t & 31) * 16;
        v8f acc = {};
        const float* arow = hbuf + (size_t)(mbase + l15) * H_;
        const float* brow = W2   + (size_t)(nbase + l15) * H_;
        for (int k0 = 0; k0 < H_; k0 += 4) {
            v2f a = *(const v2f*)(arow + k0 + kb);
            v2f b = *(const v2f*)(brow + k0 + kb);
            acc = __builtin_amdgcn_wmma_f32_16x16x4_f32(
                false, a, false, b, (short)0, acc, false, false);
        }
        const float bias = b2[nbase + l15];
#pragma unroll
        for (int r = 0; r < 8; ++r) {
            scores[(size_t)(mbase + r + 8 * hi) * C_ + (nbase + l15)] = acc[r] + bias;
        }
    }
}

// ---------------- Kernel 3: stable descending argsort (bitonic, per batch) ----------------
// key = (~orderable(score) << 32) | index : ascending sort => descending scores,
// ties broken by smaller index first (matches JAX's stable argsort of -scores).
__global__ void CSSA_sort_kernel(const float* __restrict__ scores, int* __restrict__ order) {
    __shared__ uint64_t keys[C_];          // 4 KB
    const int b   = blockIdx.x;
    const int tid = threadIdx.x;           // 0..511
    uint32_t u = __float_as_uint(scores[b * C_ + tid]);
    u ^= (u & 0x80000000u) ? 0xFFFFFFFFu : 0x80000000u;   // float -> ascending uint
    u = ~u;                                               // descending
    keys[tid] = ((uint64_t)u << 32) | (uint32_t)tid;
    __syncthreads();
    for (int k = 2; k <= C_; k <<= 1) {
        for (int j = k >> 1; j > 0; j >>= 1) {
            const int ixj = tid ^ j;
            if (ixj > tid) {
                const uint64_t a = keys[tid], c = keys[ixj];
                const bool up = ((tid & k) == 0);
                if (up ? (a > c) : (a < c)) { keys[tid] = c; keys[ixj] = a; }
            }
            __syncthreads();
        }
    }
    order[b * C_ + tid] = (int)(keys[tid] & 0x1FFu);
}

// ---------------- Kernel 4: fused channel gather + residual add ----------------
// out[b,c,:] = x[b,c,:] + x[b,order[b,c],:]  — pure bandwidth, float4 b128 traffic.
__global__ void CSSA_gather_add_kernel(const float* __restrict__ x,
                                       const int* __restrict__ order,
                                       float* __restrict__ out) {
    const int row = blockIdx.x;            // b*C_ + c
    const int b   = row >> 9;
    const int tid = threadIdx.x;
    const int src = order[row];
    const float4* xr = (const float4*)(x + (size_t)row * D_);
    const float4* xs = (const float4*)(x + ((size_t)b * C_ + src) * D_);
    float4*       o  = (float4*)(out + (size_t)row * D_);
#pragma unroll
    for (int i = 0; i < 4; ++i) {
        const int idx = tid + 256 * i;
        float4 a = xr[idx], c = xs[idx];
        o[idx] = make_float4(a.x + c.x, a.y + c.y, a.z + c.z, a.w + c.w);
    }
}

extern "C" void kernel_launch(void* const* d_in, const int* in_sizes, int n_in,
                              void* d_out, int out_size, void* d_ws, size_t ws_size,
                              hipStream_t stream) {
    const float* x  = (const float*)d_in[0];   // [32,512,4096]
    const float* W1 = (const float*)d_in[1];   // [128,512]
    const float* b1 = (const float*)d_in[2];   // [128]
    const float* W2 = (const float*)d_in[3];   // [512,128]
    const float* b2 = (const float*)d_in[4];   // [512]
    float* out = (float*)d_out;

    char* ws = (char*)d_ws;
    float* pooled = (float*)(ws);              // 32*512*4 = 64 KB
    float* scores = (float*)(ws + 65536);      // 64 KB
    int*   order  = (int*)  (ws + 131072);     // 64 KB

    CSSA_pool_kernel<<<B_ * C_, 256, 0, stream>>>(x, pooled);
    CSSA_mlp_kernel<<<1, 512, 0, stream>>>(pooled, W1, b1, W2, b2, scores);
    CSSA_sort_kernel<<<B_, C_, 0, stream>>>(scores, order);
    CSSA_gather_add_kernel<<<B_ * C_, 256, 0, stream>>>(x, order, out);
}